// HDNLoss_37349035606849
// MI455X (gfx1250) — compile-verified
//
#include <hip/hip_runtime.h>

#define NPART 512      // blocks in streaming kernels -> rows of partial-sum matrix
#define LDP   128      // partial matrix row stride (64 pred + 64 gt columns)
#define FPSCALE 1048576.0f   // 2^20 fixed-point scale for deterministic fp sums
#define EPS 1e-6f

typedef float v2f __attribute__((ext_vector_type(2)));
typedef float v8f __attribute__((ext_vector_type(8)));

__device__ __forceinline__ unsigned f2key(float f) {
    unsigned u = __float_as_uint(f);
    return u ^ ((u >> 31) ? 0xFFFFFFFFu : 0x80000000u);
}
__device__ __forceinline__ float key2f(unsigned k) {
    unsigned u = k ^ ((k >> 31) ? 0x80000000u : 0xFFFFFFFFu);
    return __uint_as_float(u);
}
__device__ __forceinline__ int bin_of(float g) {  // caller guarantees g > 0
    int b = (int)floorf(g * 64.0f);
    return b < 0 ? 0 : (b > 63 ? 63 : b);
}

// ---------------- zero a u32 range ----------------
__global__ void k_zero(unsigned* p, int n) {
    int i = blockIdx.x * blockDim.x + threadIdx.x;
    int s = gridDim.x * blockDim.x;
    for (; i < n; i += s) p[i] = 0u;
}

// ---------------- per-bin element counts (integer atomics, exact) ----------------
__global__ void k_count(const float* __restrict__ gt, unsigned* __restrict__ counts, int n) {
    __shared__ unsigned acc[64];
    int t = threadIdx.x;
    if (t < 64) acc[t] = 0u;
    __syncthreads();
    int i = blockIdx.x * blockDim.x + t;
    int s = gridDim.x * blockDim.x;
    for (; i < n; i += s) {
        float g = gt[i];
        if (g > 0.0f) atomicAdd(&acc[bin_of(g)], 1u);
    }
    __syncthreads();
    if (t < 64) atomicAdd(&counts[t], acc[t]);
}

// ---------------- init ranks (lower median rank) & prefixes ----------------
__global__ void k_rank(const unsigned* __restrict__ counts, unsigned* kk, unsigned* prefix) {
    int b = threadIdx.x;   // 64 threads
    if (b < 64) {
        unsigned c = counts[b];
        unsigned r = c ? (c - 1u) >> 1 : 0u;
        kk[b] = r;  kk[64 + b] = r;
        prefix[b] = 0u;  prefix[64 + b] = 0u;
    }
}

// ---------------- radix-select histogram pass (MSD, 8 bits/pass) ----------------
__global__ void k_hist(const float* __restrict__ pred, const float* __restrict__ gt,
                       const unsigned* __restrict__ prefix, unsigned* __restrict__ hist,
                       int n, int pass) {
    int shift = 24 - 8 * pass;
    int i = blockIdx.x * blockDim.x + threadIdx.x;
    int s = gridDim.x * blockDim.x;
    for (; i < n; i += s) {
        float g = gt[i];
        if (g > 0.0f) {
            int b = bin_of(g);
            unsigned kp = f2key(pred[i]);
            unsigned kg = f2key(g);
            unsigned hp = (pass == 0) ? 0u : (kp >> (shift + 8));
            unsigned hg = (pass == 0) ? 0u : (kg >> (shift + 8));
            if (hp == prefix[b])
                atomicAdd(&hist[(b << 8) + ((kp >> shift) & 0xFFu)], 1u);
            if (hg == prefix[64 + b])
                atomicAdd(&hist[((64 + b) << 8) + ((kg >> shift) & 0xFFu)], 1u);
        }
    }
}

// ---------------- pick digit per (array,bin), update prefix/rank, zero hist ----------------
__global__ void k_select(unsigned* __restrict__ hist, unsigned* __restrict__ prefix,
                         unsigned* __restrict__ kk, float* __restrict__ med, int pass) {
    int sb = blockIdx.x;           // 0..127 : array*64 + bin
    int t  = threadIdx.x;          // 0..255 : digit
    unsigned* h = hist + sb * 256;
    __shared__ unsigned sh[256];
    unsigned hv = h[t];
    sh[t] = hv;
    __syncthreads();
    for (int off = 1; off < 256; off <<= 1) {   // inclusive scan
        unsigned v = (t >= off) ? sh[t - off] : 0u;
        __syncthreads();
        sh[t] += v;
        __syncthreads();
    }
    unsigned incl = sh[t], excl = incl - hv;
    unsigned k = kk[sb];
    if (hv > 0u && excl <= k && k < incl) {     // exactly one thread matches
        unsigned np = (prefix[sb] << 8) | (unsigned)t;
        prefix[sb] = np;
        kk[sb] = k - excl;
        if (pass == 3) med[sb] = key2f(np);
    }
    h[t] = 0u;                                  // ready for next pass
}

// ---------------- mean-abs-deviation partial sums (deterministic fixed-point) ----------------
__global__ void k_dev(const float* __restrict__ pred, const float* __restrict__ gt,
                      const float* __restrict__ med, float* __restrict__ partials, int n) {
    __shared__ unsigned long long acc[128];
    int t = threadIdx.x;
    for (int j = t; j < 128; j += blockDim.x) acc[j] = 0ull;
    __syncthreads();
    int i = blockIdx.x * blockDim.x + t;
    int s = gridDim.x * blockDim.x;
    for (; i < n; i += s) {
        float g = gt[i];
        if (g > 0.0f) {
            int b = bin_of(g);
            float dp = fabsf(pred[i] - med[b]);
            float dg = fabsf(g - med[64 + b]);
            atomicAdd(&acc[b],      (unsigned long long)(dp * FPSCALE + 0.5f));
            atomicAdd(&acc[64 + b], (unsigned long long)(dg * FPSCALE + 0.5f));
        }
    }
    __syncthreads();
    for (int j = t; j < 128; j += blockDim.x)
        partials[blockIdx.x * LDP + j] = (float)acc[j] * (1.0f / FPSCALE);
}

// ---------------- WMMA cross-block column reduction ----------------
// D = ones(16x4) x B(4x16) + C  => every row of D accumulates column sums of B.
// Deterministic fp32 chain (fixed row order). One wave per 16-column tile.
__global__ void k_reduce_wmma(const float* __restrict__ P, float* __restrict__ out, int nrows) {
    int lane = threadIdx.x;                 // 0..31, one wave
    int col  = blockIdx.x * 16 + (lane & 15);
    int rbase = (lane >= 16) ? 2 : 0;
    v2f a; a.x = 1.0f; a.y = 1.0f;          // all-ones A (layout-invariant)
    v8f c = {0.f, 0.f, 0.f, 0.f, 0.f, 0.f, 0.f, 0.f};
    for (int r = 0; r < nrows; r += 4) {    // nrows % 4 == 0
        v2f b;
        b.x = P[(r + rbase + 0) * LDP + col];
        b.y = P[(r + rbase + 1) * LDP + col];
        c = __builtin_amdgcn_wmma_f32_16x16x4_f32(
                /*neg_a=*/false, a, /*neg_b=*/false, b,
                /*c_mod=*/(short)0, c, /*reuse_a=*/false, /*reuse_b=*/false);
    }
    if (lane < 16) out[col] = c[0];         // D row M=0 : lanes 0..15 = N 0..15
}

// ---------------- s = per-bin mean abs deviation ----------------
__global__ void k_scale(const float* __restrict__ sums, const unsigned* __restrict__ counts,
                        float* __restrict__ sbuf) {
    int j = threadIdx.x;   // 128 threads
    if (j < 128) sbuf[j] = sums[j] / fmaxf((float)counts[j & 63], 1.0f);
}

// ---------------- normalized-diff partial sums ----------------
__global__ void k_diff(const float* __restrict__ pred, const float* __restrict__ gt,
                       const float* __restrict__ med, const float* __restrict__ sbuf,
                       float* __restrict__ partials, int n) {
    __shared__ unsigned long long acc[64];
    int t = threadIdx.x;
    for (int j = t; j < 64; j += blockDim.x) acc[j] = 0ull;
    __syncthreads();
    int i = blockIdx.x * blockDim.x + t;
    int s = gridDim.x * blockDim.x;
    for (; i < n; i += s) {
        float g = gt[i];
        if (g > 0.0f) {
            int b = bin_of(g);
            float np = (pred[i] - med[b])      / (sbuf[b]      + EPS);
            float ng = (g       - med[64 + b]) / (sbuf[64 + b] + EPS);
            float d = fabsf(np - ng);
            atomicAdd(&acc[b], (unsigned long long)(d * FPSCALE + 0.5f));
        }
    }
    __syncthreads();
    for (int j = t; j < 64; j += blockDim.x)
        partials[blockIdx.x * LDP + j] = (float)acc[j] * (1.0f / FPSCALE);
}

// ---------------- final scalar ----------------
__global__ void k_final(const float* __restrict__ per_bin, const unsigned* __restrict__ counts,
                        float* __restrict__ out) {
    if (threadIdx.x == 0 && blockIdx.x == 0) {
        float acc = 0.0f;
        for (int b = 0; b < 64; ++b) {
            unsigned c = counts[b];
            if (c > 0u) acc += per_bin[b] / (float)c;
        }
        out[0] = acc / 64.0f;
    }
}

extern "C" void kernel_launch(void* const* d_in, const int* in_sizes, int n_in,
                              void* d_out, int out_size, void* d_ws, size_t ws_size,
                              hipStream_t stream) {
    const float* pred = (const float*)d_in[0];
    const float* gt   = (const float*)d_in[1];
    (void)d_in; (void)n_in; (void)out_size; (void)ws_size;   // num_bins fixed at 64
    int n = in_sizes[0];
    float* out = (float*)d_out;

    // workspace layout (u32 units)
    unsigned* ws      = (unsigned*)d_ws;
    unsigned* counts  = ws;                    //   64
    unsigned* kk      = ws + 64;               //  128
    unsigned* prefix  = ws + 192;              //  128
    float*    med     = (float*)(ws + 320);    //  128
    float*    sbuf    = (float*)(ws + 448);    //  128
    float*    sums    = (float*)(ws + 576);    //  128
    unsigned* hist    = ws + 704;              //  2*64*256 = 32768
    float*    partials= (float*)(ws + 33472);  //  NPART*LDP = 512*128 floats

    k_zero<<<131, 256, 0, stream>>>(ws, 33472);                    // counts..hist
    k_count<<<NPART, 256, 0, stream>>>(gt, counts, n);
    k_rank<<<1, 64, 0, stream>>>(counts, kk, prefix);

    for (int p = 0; p < 4; ++p) {                                  // exact per-bin medians
        k_hist<<<NPART, 256, 0, stream>>>(pred, gt, prefix, hist, n, p);
        k_select<<<128, 256, 0, stream>>>(hist, prefix, kk, med, p);
    }

    k_dev<<<NPART, 256, 0, stream>>>(pred, gt, med, partials, n);
    k_reduce_wmma<<<8, 32, 0, stream>>>(partials, sums, NPART);    // 128 cols (pred+gt)
    k_scale<<<1, 128, 0, stream>>>(sums, counts, sbuf);

    k_diff<<<NPART, 256, 0, stream>>>(pred, gt, med, sbuf, partials, n);
    k_reduce_wmma<<<4, 32, 0, stream>>>(partials, sums, NPART);    // 64 cols
    k_final<<<1, 64, 0, stream>>>(sums, counts, out);
}